// Drop_Edge_68032281969089
// MI455X (gfx1250) — compile-verified
//
#include <hip/hip_runtime.h>
#include <stdint.h>

// Native 4xf32 vector so __builtin_nontemporal_{load,store} accept it.
typedef float v4f __attribute__((ext_vector_type(4)));

// floor(0.7 * 65536): per-16-bit-field keep threshold for Bernoulli(keep=0.7).
#define KEEP_THRESH 45875u
#define SEED 0x9E3779B9u

__device__ __forceinline__ unsigned mix32(unsigned x) {
    // xorshift-multiply finalizer (lowbias32): cheap, good avalanche.
    x ^= x >> 16; x *= 0x7feb352du;
    x ^= x >> 15; x *= 0x846ca68bu;
    x ^= x >> 16;
    return x;
}

__device__ __forceinline__ v4f drop_mask(v4f v, unsigned g4) {
    // Four 16-bit uniforms from two 32-bit hashes of the element indices.
    const unsigned h0 = mix32((g4 * 2u + 0u) ^ SEED);
    const unsigned h1 = mix32((g4 * 2u + 1u) ^ SEED);
    v.x = ((h0 & 0xFFFFu) < KEEP_THRESH) ? v.x : 0.0f;
    v.y = ((h0 >> 16)     < KEEP_THRESH) ? v.y : 0.0f;
    v.z = ((h1 & 0xFFFFu) < KEEP_THRESH) ? v.z : 0.0f;
    v.w = ((h1 >> 16)     < KEEP_THRESH) ? v.w : 0.0f;
    return v;
}

// ---------------------------------------------------------------------------
// adj drop-edge: double-buffered async global->LDS pipeline, 2 slots/thread.
// Tile = 512 float4 = 8 KB (256 threads x 2 slots). Each thread async-loads
// its two 16 B slots for the NEXT tile (second slot via offset:4096, which the
// hardware adds to BOTH the LDS and global addresses), then s_wait_asynccnt 2
// leaves the next tile's pair outstanding while guaranteeing the current tile
// is resident (per-wave in-order async completion). Threads only read the LDS
// bytes they themselves requested -> no workgroup barrier needed.
// ---------------------------------------------------------------------------
__global__ void __launch_bounds__(256)
drop_edge_kernel(const float* __restrict__ adj, float* __restrict__ out,
                 unsigned num_tiles) {
    __shared__ v4f lbuf[2][512];  // 16 KB double buffer

    const unsigned tid = threadIdx.x;
    const unsigned stride = gridDim.x;
    const v4f* __restrict__ adj4 = (const v4f*)adj;
    v4f* __restrict__ out4 = (v4f*)out;

    // Workgroup-relative LDS byte offsets of this thread's slot pair base
    // (flat LDS address low 32 bits == addrspace(3) offset).
    const unsigned lds0 = (unsigned)(unsigned long long)(&lbuf[0][tid]);
    const unsigned lds1 = (unsigned)(unsigned long long)(&lbuf[1][tid]);

    unsigned t = blockIdx.x;
    if (t >= num_tiles) return;

    // Prologue: kick off tile t (both slots) into buffer 0.
    {
        const v4f* g = adj4 + ((unsigned long long)t * 512ull + tid);
        asm volatile("global_load_async_to_lds_b128 %0, %1, off\n\t"
                     "global_load_async_to_lds_b128 %0, %1, off offset:4096"
                     :: "v"(lds0), "v"(g) : "memory");
    }

    unsigned iter = 0;
    for (; t < num_tiles; t += stride, ++iter) {
        const unsigned nt = t + stride;
        const unsigned cur = iter & 1u;

        if (nt < num_tiles) {
            // Issue next tile into the other buffer, then wait for current.
            const unsigned lds_next = cur ? lds0 : lds1;
            const v4f* g = adj4 + ((unsigned long long)nt * 512ull + tid);
            asm volatile("global_load_async_to_lds_b128 %0, %1, off\n\t"
                         "global_load_async_to_lds_b128 %0, %1, off offset:4096"
                         :: "v"(lds_next), "v"(g) : "memory");
            asm volatile("s_wait_asynccnt 0x2" ::: "memory");
        } else {
            asm volatile("s_wait_asynccnt 0x0" ::: "memory");
        }

        // ds_load_b128 of this thread's own two slots.
        v4f a = lbuf[cur][tid];
        v4f b = lbuf[cur][tid + 256u];

        const unsigned ga = t * 512u + tid;          // float4 index (< 2^24)
        const unsigned gb = ga + 256u;
        a = drop_mask(a, ga);
        b = drop_mask(b, gb);

        __builtin_nontemporal_store(a, out4 + ga);
        __builtin_nontemporal_store(b, out4 + gb);
    }
}

// ---------------------------------------------------------------------------
// x passthrough: 16 MB streaming copy, B128 + non-temporal both directions,
// unrolled x4 so four loads clause together and overlap their latency.
// ---------------------------------------------------------------------------
__global__ void __launch_bounds__(256)
copy_x_kernel(const float* __restrict__ x, float* __restrict__ out, unsigned n4) {
    const v4f* __restrict__ xin = (const v4f*)x;
    v4f* __restrict__ o = (v4f*)out;
    const unsigned T = gridDim.x * 256u;
    unsigned i = blockIdx.x * 256u + threadIdx.x;
    for (; i + 3u * T < n4; i += 4u * T) {
        v4f v0 = __builtin_nontemporal_load(xin + i);
        v4f v1 = __builtin_nontemporal_load(xin + i + T);
        v4f v2 = __builtin_nontemporal_load(xin + i + 2u * T);
        v4f v3 = __builtin_nontemporal_load(xin + i + 3u * T);
        __builtin_nontemporal_store(v0, o + i);
        __builtin_nontemporal_store(v1, o + i + T);
        __builtin_nontemporal_store(v2, o + i + 2u * T);
        __builtin_nontemporal_store(v3, o + i + 3u * T);
    }
    for (; i < n4; i += T) {
        v4f v = __builtin_nontemporal_load(xin + i);
        __builtin_nontemporal_store(v, o + i);
    }
}

// ---------------------------------------------------------------------------
extern "C" void kernel_launch(void* const* d_in, const int* in_sizes, int n_in,
                              void* d_out, int out_size, void* d_ws, size_t ws_size,
                              hipStream_t stream) {
    (void)n_in; (void)out_size; (void)d_ws; (void)ws_size;

    const float* x   = (const float*)d_in[0];
    const float* adj = (const float*)d_in[1];
    float* out = (float*)d_out;

    const unsigned xn = (unsigned)in_sizes[0];   // 8192*512  = 4,194,304
    const unsigned an = (unsigned)in_sizes[1];   // 8192*8192 = 67,108,864

    // Output layout: [x | t], flat in return order.
    const unsigned x4 = xn / 4u;                 // 1,048,576 float4
    const unsigned a4 = an / 4u;                 // 16,777,216 float4
    const unsigned num_tiles = a4 / 512u;        // 32,768 tiles of 8 KB

    copy_x_kernel<<<1024, 256, 0, stream>>>(x, out, x4);
    drop_edge_kernel<<<4096, 256, 0, stream>>>(adj, out + xn, num_tiles);
}